// VectorQuantizer_55052890800159
// MI455X (gfx1250) — compile-verified
//
#include <hip/hip_runtime.h>
#include <stdint.h>

// ---------------------------------------------------------------------------
// VQ nearest-code search on MI455X (gfx1250):
//   distances = ||e_k||^2 - 2 * X.E^T   (||x||^2 dropped: row-constant)
//   GEMM: V_WMMA_F32_16X16X4_F32 (fp32 needed for argmin margins).
//   Staging: GLOBAL_LOAD_ASYNC_TO_LDS_B128 (ASYNCcnt) + double-buffered LDS.
//   Argmin: lane-local running min (8 packed u64/lane), cross-lane reduce and
//   ds_min_u64 commit once at the end -> hot loop is WMMA + DS frags only.
// 137.4 GFLOP GEMM vs ~84MB traffic -> compute bound; codebook lives in L2.
// ---------------------------------------------------------------------------

typedef float v2f __attribute__((ext_vector_type(2)));
typedef float v8f __attribute__((ext_vector_type(8)));

static constexpr int DIM        = 512;
static constexpr int KCODES     = 8192;
static constexpr int ROWS_PER_B = 32;    // rows per workgroup (2 row tiles)
static constexpr int KTILE      = 64;    // codes per K iteration (4 col tiles)
static constexpr int DCHUNK     = 64;    // D staged per LDS chunk
static constexpr int XLD        = 68;    // padded LDS stride (68 mod 64 = 4 banks)
static constexpr int ELD        = 68;
static constexpr int NCH        = (KCODES / KTILE) * (DIM / DCHUNK);  // 1024

// async DMA global->LDS, 16B per lane; IOFFSET applies to BOTH addresses
#define ASYNC_B128(ldsoff, gptr, OFF)                                          \
    asm volatile("global_load_async_to_lds_b128 %0, %1, off offset:" #OFF     \
                 :: "v"(ldsoff), "v"(gptr) : "memory")

__device__ __forceinline__ unsigned int fkey(float f) {
    // monotone float -> u32 map (ascending order preserved, incl. negatives)
    unsigned int u = __float_as_uint(f);
    return u ^ ((unsigned int)((int)u >> 31) | 0x80000000u);
}

// ---------------------------------------------------------------- Kernel A --
__global__ void vq_norms_kernel(const float* __restrict__ emb,
                                float* __restrict__ e_norms,
                                float* __restrict__ used) {
    int k = blockIdx.x * blockDim.x + threadIdx.x;
    if (k < KCODES) {
        const float4* row = (const float4*)(emb + (size_t)k * DIM);
        float s = 0.f;
        #pragma unroll 8
        for (int i = 0; i < DIM / 4; ++i) {
            float4 v = row[i];
            s += v.x * v.x + v.y * v.y + v.z * v.z + v.w * v.w;
        }
        e_norms[k] = s;
        used[k]    = 0.f;
    }
}

// ---------------------------------------------------------------- Kernel B --
__global__ __launch_bounds__(256)
void vq_main_kernel(const float* __restrict__ x,
                    const float* __restrict__ emb,
                    const float* __restrict__ e_norms,
                    float* __restrict__ used,
                    float* __restrict__ out_q,
                    float* __restrict__ out_idx,
                    float* __restrict__ partials) {
    __shared__ float Xc[2][ROWS_PER_B * XLD];       // 2 x  8.7 KB
    __shared__ float Ec[2][KTILE * ELD];            // 2 x 17.4 KB
    __shared__ unsigned long long minTab[ROWS_PER_B];
    __shared__ float rbuf[256];

    const int t        = threadIdx.x;
    const int lane     = t & 31;                    // wave32
    const int wave     = t >> 5;                    // 0..7
    const int row_tile = wave >> 2;                 // 0..1
    const int col_tile = wave & 3;                  // 0..3
    const int m        = lane & 15;                 // A row / B col within tile
    const int kb       = (lane >> 4) << 1;          // K sub-offset {0,2}
    const long gr0     = (long)blockIdx.x * ROWS_PER_B;

    if (t < ROWS_PER_B) minTab[t] = ~0ull;

    // staging assignments
    const int xr = t >> 3, xj = t & 7;              // X: row, 8-float slice
    const int er = t >> 2, eq = t & 3;              // E: row, 16-float slice

    // per-thread LDS byte offsets (addrspace(3) offset == low 32 addr bits)
    const uint32_t xl[2] = {
        (uint32_t)(uintptr_t)&Xc[0][xr * XLD + xj * 8],
        (uint32_t)(uintptr_t)&Xc[1][xr * XLD + xj * 8] };
    const uint32_t el[2] = {
        (uint32_t)(uintptr_t)&Ec[0][er * ELD + eq * 16],
        (uint32_t)(uintptr_t)&Ec[1][er * ELD + eq * 16] };

    const float* xg_base = x + (gr0 + xr) * DIM + xj * 8;   // advance 64/chunk
    const float* eg_base = emb + (size_t)er * DIM + eq * 16;

    // prologue: stream chunk 0 into buffer 0
    {
        const float* xga = xg_base;
        const float* ega = eg_base;
        ASYNC_B128(xl[0], xga, 0);  ASYNC_B128(xl[0], xga, 16);
        ASYNC_B128(el[0], ega, 0);  ASYNC_B128(el[0], ega, 16);
        ASYNC_B128(el[0], ega, 32); ASYNC_B128(el[0], ega, 48);
    }

    v8f acc = {0.f, 0.f, 0.f, 0.f, 0.f, 0.f, 0.f, 0.f};
    unsigned long long pkmin[8];
    #pragma unroll
    for (int i = 0; i < 8; ++i) pkmin[i] = ~0ull;   // lane-local running argmin

    for (int g = 0; g < NCH; ++g) {
        const int b  = g & 1;
        const int c  = g & 7;                 // d-chunk within K-tile
        const int kt = (g >> 3) * KTILE;      // K-tile base

        __syncthreads();                      // buf[(g+1)&1] free to overwrite

        if (g + 1 < NCH) {                    // stream chunk g+1 into buf[!b]
            const int gn = g + 1;
            const float* xga = xg_base + (gn & 7) * DCHUNK;
            const float* ega = eg_base + (size_t)((gn >> 3) * KTILE) * DIM
                                       + (gn & 7) * DCHUNK;
            const uint32_t xln = xl[gn & 1], eln = el[gn & 1];
            ASYNC_B128(xln, xga, 0);  ASYNC_B128(xln, xga, 16);
            ASYNC_B128(eln, ega, 0);  ASYNC_B128(eln, ega, 16);
            ASYNC_B128(eln, ega, 32); ASYNC_B128(eln, ega, 48);
            if (g + 2 < NCH)                  // warm L2 for chunk g+2 (E)
                __builtin_prefetch(eg_base + (size_t)(((g + 2) >> 3) * KTILE) * DIM
                                           + ((g + 2) & 7) * DCHUNK, 0, 3);
            asm volatile("s_wait_asynccnt 0x6" ::: "memory");   // batch g landed
        } else {
            asm volatile("s_wait_asynccnt 0x0" ::: "memory");
        }
        __syncthreads();                      // all waves' batch g visible

        if (c == 0)
            acc = (v8f){0.f, 0.f, 0.f, 0.f, 0.f, 0.f, 0.f, 0.f};

        const float* arow = &Xc[b][(row_tile * 16 + m) * XLD + kb];
        const float* brow = &Ec[b][(col_tile * 16 + m) * ELD + kb];
        #pragma unroll
        for (int s = 0; s < DCHUNK / 4; ++s) {
            v2f a = *(const v2f*)(arow + 4 * s);
            v2f bb = *(const v2f*)(brow + 4 * s);
            acc = __builtin_amdgcn_wmma_f32_16x16x4_f32(
                      false, a, false, bb, (short)0, acc, false, false);
        }

        if (c == 7) {
            // dist = ||e||^2 - 2*dot ; fold into lane-local running argmin
            const int kglob = kt + col_tile * 16 + m;
            const float en  = e_norms[kglob];
            #pragma unroll
            for (int i = 0; i < 8; ++i) {
                float dist = en - 2.0f * acc[i];
                unsigned long long pk =
                    ((unsigned long long)fkey(dist) << 32) | (unsigned int)kglob;
                if (pk < pkmin[i]) pkmin[i] = pk;
            }
        }
    }

    // one-time cross-lane reduce: xor masks 1..8 stay inside each 16-lane half
    #pragma unroll
    for (int off = 1; off < 16; off <<= 1) {
        #pragma unroll
        for (int i = 0; i < 8; ++i) {
            unsigned long long o = __shfl_xor(pkmin[i], off);
            if (o < pkmin[i]) pkmin[i] = o;
        }
    }
    // lanes (L&15)<8 commit one row each via ds_min_u64 (combines the 4 col waves)
    if ((lane & 15) < 8) {
        int slot = lane & 7;
        int rloc = row_tile * 16 + ((lane >> 4) << 3) + slot;
        atomicMin(&minTab[rloc], pkmin[slot]);
    }
    __syncthreads();

    // ---- epilogue: indices, usage flags, gather + straight-through output --
    if (t < ROWS_PER_B) {
        unsigned int idx = (unsigned int)(minTab[t] & 0xffffffffull);
        out_idx[gr0 + t] = (float)idx;
        used[idx] = 1.0f;
    }
    __syncthreads();

    const int qr = t >> 3;          // row 0..31
    const int qc = t & 7;           // 64-dim slice
    unsigned int idxr = (unsigned int)(minTab[qr] & 0xffffffffull);
    const float4* qrow = (const float4*)(emb + (size_t)idxr * DIM + qc * 64);
    const float4* xrow = (const float4*)(x + (gr0 + qr) * DIM + qc * 64);
    float4* orow = (float4*)(out_q + (gr0 + qr) * DIM + qc * 64);
    float ls = 0.f;
    #pragma unroll
    for (int i = 0; i < 16; ++i) {
        float4 q = qrow[i], xv = xrow[i];
        float dx = q.x - xv.x, dy = q.y - xv.y, dz = q.z - xv.z, dw = q.w - xv.w;
        float4 o;                      // match reference: x + (q - x)
        o.x = xv.x + dx; o.y = xv.y + dy; o.z = xv.z + dz; o.w = xv.w + dw;
        orow[i] = o;
        ls += dx * dx + dy * dy + dz * dz + dw * dw;
    }
    rbuf[t] = ls;
    __syncthreads();
    for (int s = 128; s > 0; s >>= 1) {
        if (t < s) rbuf[t] += rbuf[t + s];
        __syncthreads();
    }
    if (t == 0) partials[blockIdx.x] = rbuf[0];
}

// ---------------------------------------------------------------- Kernel C --
__global__ void vq_finalize_kernel(const float* __restrict__ partials, int nPart,
                                   const float* __restrict__ used,
                                   float* __restrict__ out_scalars, float invN) {
    __shared__ float rb[256];
    int t = threadIdx.x;
    float s = 0.f;
    for (int i = t; i < nPart; i += 256) s += partials[i];
    rb[t] = s;
    __syncthreads();
    for (int sft = 128; sft > 0; sft >>= 1) {
        if (t < sft) rb[t] += rb[t + sft];
        __syncthreads();
    }
    float total = rb[0];
    __syncthreads();
    float c = 0.f;
    for (int i = t; i < KCODES; i += 256) c += (used[i] != 0.f) ? 1.f : 0.f;
    rb[t] = c;
    __syncthreads();
    for (int sft = 128; sft > 0; sft >>= 1) {
        if (t < sft) rb[t] += rb[t + sft];
        __syncthreads();
    }
    if (t == 0) {
        float mse = total * invN;
        out_scalars[0] = 0.25f * mse;               // vq_loss (BETA)
        out_scalars[1] = mse;                       // e_latent_loss
        out_scalars[2] = rb[0] / (float)KCODES;     // codebook_usage
    }
}

// ------------------------------------------------------------------ launch --
extern "C" void kernel_launch(void* const* d_in, const int* in_sizes, int n_in,
                              void* d_out, int out_size, void* d_ws, size_t ws_size,
                              hipStream_t stream) {
    const float* x   = (const float*)d_in[0];   // [4,4096,512] fp32
    const float* emb = (const float*)d_in[1];   // [8192,512]   fp32
    float* out = (float*)d_out;

    const int N = in_sizes[0] / DIM;            // 16384 rows
    float* ws       = (float*)d_ws;
    float* e_norms  = ws;                       // [8192]
    float* used     = ws + KCODES;              // [8192]
    float* partials = ws + 2 * KCODES;          // [N/32]

    float* out_q       = out;                           // [N*512]
    float* out_idx     = out + (size_t)N * DIM;         // [N]
    float* out_scalars = out + (size_t)N * DIM + N;     // [3]

    const int nBlocksB = N / ROWS_PER_B;

    vq_norms_kernel<<<(KCODES + 255) / 256, 256, 0, stream>>>(emb, e_norms, used);
    vq_main_kernel<<<nBlocksB, 256, 0, stream>>>(x, emb, e_norms, used,
                                                 out_q, out_idx, partials);
    vq_finalize_kernel<<<1, 256, 0, stream>>>(partials, nBlocksB, used, out_scalars,
                                              1.0f / ((float)N * (float)DIM));
}